// AttnLoss_84224308674705
// MI455X (gfx1250) — compile-verified
//
#include <hip/hip_runtime.h>
#include <hip/hip_bf16.h>
#include <math.h>

// ---------------------------------------------------------------------------
// Problem geometry (compile-time constants from the reference)
// ---------------------------------------------------------------------------
#define BDIM 16
#define TDIM 8
#define CDIM 64
#define PDIM 2048
#define NROWS (BDIM * TDIM * CDIM)       // 8192 rows of PDIM floats (8 KB each)
#define ROWS_PER_BLOCK 8
#define NBLOCKS (NROWS / ROWS_PER_BLOCK) // 1024 (multiple of 64 -> clean WMMA chunks)
#define THREADS 256
#define NTOT ((float)(BDIM * TDIM * CDIM * PDIM))

typedef unsigned int v4u_t __attribute__((ext_vector_type(4)));
typedef int          v8i_t __attribute__((ext_vector_type(8)));
typedef int          v4i_t __attribute__((ext_vector_type(4)));
typedef float        v2f_t __attribute__((ext_vector_type(2)));
typedef float        v8f_t __attribute__((ext_vector_type(8)));

// ---------------------------------------------------------------------------
// LDS layout for the main kernel: double-buffered 3 source rows + pP tables
// ---------------------------------------------------------------------------
struct Smem {
    float buf[2][3][PDIM];   // 48 KB : TDM destination tiles
    int   pP[3][PDIM];       // 24 KB : in-row permutation tables
    float red[4][THREADS / 32];
};

// Build a D# for a 1-D row copy (PDIM f32, contiguous) and issue TDM load.
// D# group0: count=1 | lds_addr | global_addr | type=2
// D# group1: data_size=4B, tensor_dim0=tile_dim0=PDIM, tensor_dim1=tile_dim1=1,
//            tensor_dim0_stride=PDIM. Groups 2/3 unused (zeros).
__device__ __forceinline__ void tdm_row_load(const float* gsrc, unsigned lds_off) {
    unsigned long long ga = (unsigned long long)(uintptr_t)gsrc;
    v4u_t g0;
    g0[0] = 1u;                                               // count=1, no gather
    g0[1] = lds_off;                                          // lds_addr (bytes)
    g0[2] = (unsigned)(ga & 0xffffffffull);                   // global_addr[31:0]
    g0[3] = (unsigned)((ga >> 32) & 0x1ffffffull) | (2u << 30); // addr[56:32] | type=2
    v8i_t g1;
    g1[0] = (int)(2u << 16);                                  // data_size = 4 bytes
    g1[1] = (int)(((unsigned)PDIM & 0xffffu) << 16);          // tensor_dim0[15:0]
    g1[2] = (int)(((unsigned)PDIM >> 16) | (1u << 16));       // dim0 hi | tensor_dim1=1
    g1[3] = (int)(((unsigned)PDIM) << 16);                    // tile_dim0 = PDIM
    g1[4] = 1;                                                // tile_dim1 = 1
    g1[5] = PDIM;                                             // tensor_dim0_stride
    g1[6] = 0;
    g1[7] = 0;
    v4i_t gz4 = {0, 0, 0, 0};
    v8i_t gz8 = {0, 0, 0, 0, 0, 0, 0, 0};
    // 6-arg form (clang-23 / therock headers): two unused 128-bit groups,
    // one unused 256-bit group, cache-policy immediate.
    __builtin_amdgcn_tensor_load_to_lds(g0, g1, gz4, gz4, gz8, 0); // TENSORcnt++
}

// ---------------------------------------------------------------------------
// Main kernel: per-block partial sums of the four reductions.
// ---------------------------------------------------------------------------
__global__ __launch_bounds__(THREADS) void AttnLoss_main(
    const float* __restrict__ x, const float* __restrict__ attn,
    const float* __restrict__ noise, const unsigned char* __restrict__ mask,
    const int* __restrict__ pB1, const int* __restrict__ pT1,
    const int* __restrict__ pC1, const int* __restrict__ pP1,
    const int* __restrict__ pB2, const int* __restrict__ pT2,
    const int* __restrict__ pC2, const int* __restrict__ pP2,
    const int* __restrict__ pB3, const int* __restrict__ pT3,
    const int* __restrict__ pC3, const int* __restrict__ pP3,
    float* __restrict__ ws)
{
    __shared__ Smem s;
    const int tid = threadIdx.x;
    const bool wave0 = (tid < 32);

    // Cooperative load of the in-row permutation tables into LDS.
    for (int i = tid; i < PDIM; i += THREADS) {
        s.pP[0][i] = pP1[i];
        s.pP[1][i] = pP2[i];
        s.pP[2][i] = pP3[i];
    }

    const int row0 = blockIdx.x * ROWS_PER_BLOCK;

    // Issue the 3 TDM source-row copies for `row` into buffer `bi`.
    auto issue = [&](int row, int bi) {
        const int c = row % CDIM;
        const int t = (row / CDIM) % TDIM;
        const int b = row / (CDIM * TDIM);
        const int sr1 = (pB1[b] * TDIM + pT1[t]) * CDIM + pC1[c];
        const int sr2 = (pB2[b] * TDIM + pT2[t]) * CDIM + pC2[c];
        const int sr3 = (pB3[b] * TDIM + pT3[t]) * CDIM + pC3[c];
        tdm_row_load(x + (size_t)sr1 * PDIM, (unsigned)(uintptr_t)&s.buf[bi][0][0]);
        tdm_row_load(x + (size_t)sr2 * PDIM, (unsigned)(uintptr_t)&s.buf[bi][1][0]);
        tdm_row_load(x + (size_t)sr3 * PDIM, (unsigned)(uintptr_t)&s.buf[bi][2][0]);
    };

    if (wave0) {
        issue(row0, 0);
        __builtin_amdgcn_s_wait_tensorcnt(0);
    }
    __syncthreads();   // pP tables + first buffer ready

    float sp = 0.f, s1 = 0.f, s2 = 0.f, s3 = 0.f;

    for (int i = 0; i < ROWS_PER_BLOCK; ++i) {
        const int bi = i & 1;
        // Prefetch next row's source tiles into the other buffer (wave 0).
        if (wave0 && (i + 1) < ROWS_PER_BLOCK) issue(row0 + i + 1, bi ^ 1);

        const size_t base = (size_t)(row0 + i) * PDIM;
        const int p0 = tid * 8;                 // 8 consecutive elements / thread

        const float4 xa = *(const float4*)(x + base + p0);
        const float4 xb = *(const float4*)(x + base + p0 + 4);
        const float4 aa = *(const float4*)(attn + base + p0);
        const float4 ab = *(const float4*)(attn + base + p0 + 4);
        const float4 na = *(const float4*)(noise + base + p0);
        const float4 nb = *(const float4*)(noise + base + p0 + 4);
        const uint2  mv = *(const uint2*)(mask + base + p0);

        float xv[8] = {xa.x, xa.y, xa.z, xa.w, xb.x, xb.y, xb.z, xb.w};
        float av[8] = {aa.x, aa.y, aa.z, aa.w, ab.x, ab.y, ab.z, ab.w};
        float nv[8] = {na.x, na.y, na.z, na.w, nb.x, nb.y, nb.z, nb.w};

        #pragma unroll
        for (int u = 0; u < 8; ++u) {
            const unsigned mb = (u < 4) ? ((mv.x >> (8 * u)) & 0xffu)
                                        : ((mv.y >> (8 * (u - 4))) & 0xffu);
            if (mb) sp += av[u] * nv[u] * nv[u];

            const int pe = p0 + u;
            const float g1v = s.buf[bi][0][s.pP[0][pe]];
            const float g2v = s.buf[bi][1][s.pP[1][pe]];
            const float g3v = s.buf[bi][2][s.pP[2][pe]];
            const float d1 = xv[u] - g1v;
            const float d2 = xv[u] - g2v;
            const float d3 = xv[u] - g3v;
            s1 += av[u] * d1 * d1;
            s2 += av[u] * d2 * d2;
            s3 += av[u] * d3 * d3;
        }

        if (wave0 && (i + 1) < ROWS_PER_BLOCK) __builtin_amdgcn_s_wait_tensorcnt(0);
        __syncthreads();  // next buffer filled; current buffer free for reuse
    }

    // Block reduction: wave shfl, then across the 8 waves via LDS.
    #pragma unroll
    for (int off = 16; off > 0; off >>= 1) {
        sp += __shfl_down(sp, off, 32);
        s1 += __shfl_down(s1, off, 32);
        s2 += __shfl_down(s2, off, 32);
        s3 += __shfl_down(s3, off, 32);
    }
    const int wave = tid >> 5;
    if ((tid & 31) == 0) {
        s.red[0][wave] = sp; s.red[1][wave] = s1;
        s.red[2][wave] = s2; s.red[3][wave] = s3;
    }
    __syncthreads();
    if (tid == 0) {
        float r0 = 0.f, r1 = 0.f, r2 = 0.f, r3 = 0.f;
        #pragma unroll
        for (int w = 0; w < THREADS / 32; ++w) {
            r0 += s.red[0][w]; r1 += s.red[1][w];
            r2 += s.red[2][w]; r3 += s.red[3][w];
        }
        ws[0 * NBLOCKS + blockIdx.x] = r0;
        ws[1 * NBLOCKS + blockIdx.x] = r1;
        ws[2 * NBLOCKS + blockIdx.x] = r2;
        ws[3 * NBLOCKS + blockIdx.x] = r3;
    }
}

// ---------------------------------------------------------------------------
// Final reduction: one wave. D = ones(16x4) x B(4x16) + C column-sums 64
// per-block partials per V_WMMA_F32_16X16X4_F32 issue, accumulating in f32 C.
// B layout (4x16, 2 VGPRs): lane l<16 -> B[0][l],B[1][l]; lane l+16 -> B[2][l],B[3][l].
// ---------------------------------------------------------------------------
__global__ __launch_bounds__(32) void AttnLoss_final(const float* __restrict__ ws,
                                                     float* __restrict__ out)
{
    const int lane = threadIdx.x;
    const int col  = lane & 15;
    const int hi   = lane >> 4;

    v2f_t ones; ones[0] = 1.0f; ones[1] = 1.0f;
    float S[4];

    #pragma unroll
    for (int q = 0; q < 4; ++q) {
        const float* w = ws + q * NBLOCKS;
        v8f_t acc = {0.f, 0.f, 0.f, 0.f, 0.f, 0.f, 0.f, 0.f};
        for (int ch = 0; ch < NBLOCKS / 64; ++ch) {
            const int i0 = ch * 64 + col + hi * 32;
            v2f_t bv; bv[0] = w[i0]; bv[1] = w[i0 + 16];
            // D[m][n] = sum_k B[k][n] + C[m][n]  (A = ones)
            acc = __builtin_amdgcn_wmma_f32_16x16x4_f32(
                false, ones, false, bv, (short)0, acc, false, false);
        }
        float sum = acc[0];     // every lane: column sum for N = lane & 15
        #pragma unroll
        for (int m = 8; m >= 1; m >>= 1) sum += __shfl_xor(sum, m, 32);
        S[q] = sum;
    }

    if (lane == 0) {
        const float invN = 1.0f / NTOT;
        const float lp = S[0] * invN;
        const float l1 = S[1] * invN;
        const float l2 = S[2] * invN;
        const float l3 = S[3] * invN;
        out[0] = -lp + logf(expf(l1) + expf(l2) + expf(l3));
    }
}

// ---------------------------------------------------------------------------
// Host entry
// ---------------------------------------------------------------------------
extern "C" void kernel_launch(void* const* d_in, const int* in_sizes, int n_in,
                              void* d_out, int out_size, void* d_ws, size_t ws_size,
                              hipStream_t stream) {
    const float*         x     = (const float*)d_in[0];
    const float*         attn  = (const float*)d_in[1];
    const float*         noise = (const float*)d_in[2];
    const unsigned char* mask  = (const unsigned char*)d_in[3];
    const int* pB1 = (const int*)d_in[4];
    const int* pT1 = (const int*)d_in[5];
    const int* pC1 = (const int*)d_in[6];
    const int* pP1 = (const int*)d_in[7];
    const int* pB2 = (const int*)d_in[8];
    const int* pT2 = (const int*)d_in[9];
    const int* pC2 = (const int*)d_in[10];
    const int* pP2 = (const int*)d_in[11];
    const int* pB3 = (const int*)d_in[12];
    const int* pT3 = (const int*)d_in[13];
    const int* pC3 = (const int*)d_in[14];
    const int* pP3 = (const int*)d_in[15];

    float* ws  = (float*)d_ws;     // 4 * NBLOCKS floats, fully rewritten each call
    float* out = (float*)d_out;

    AttnLoss_main<<<NBLOCKS, THREADS, 0, stream>>>(
        x, attn, noise, mask,
        pB1, pT1, pC1, pP1,
        pB2, pT2, pC2, pP2,
        pB3, pT3, pC3, pP3, ws);
    AttnLoss_final<<<1, 32, 0, stream>>>(ws, out);
}